// StatefulFormantFilter_63565515981199
// MI455X (gfx1250) — compile-verified
//
#include <hip/hip_runtime.h>
#include <stdint.h>

// ---------------- problem constants ----------------
#define BB 32
#define TT 48000
#define FF 5
#define LL 25                 // chunk length (serial steps per thread)
#define CC (TT / LL)          // 1920 chunks per batch row
#define WCH 32                // chunks per workgroup == blockDim.x (1 wave, wave32)
#define SPAN (WCH * LL)       // 800 timesteps staged per workgroup tile
#define NBLK (CC / WCH)       // 60 blocks per batch row

#define PI_F 3.14159265358979f
#define TWO_PI_F 6.28318530717959f
#define INV_SR (1.0f / 48000.0f)

// ---------------- CDNA5 TDM availability ----------------
#if defined(__has_builtin)
#if __has_builtin(__builtin_amdgcn_tensor_load_to_lds) && \
    __has_builtin(__builtin_amdgcn_s_wait_tensorcnt)
#define USE_TDM 1
#endif
#endif

typedef __attribute__((ext_vector_type(4))) unsigned int v4u;
typedef __attribute__((ext_vector_type(8))) int v8i;
typedef __attribute__((ext_vector_type(4))) int v4i;

// Build a 1-D contiguous TDM descriptor (D#) and issue TENSOR_LOAD_TO_LDS.
// Layout per CDNA5 ISA ch.8: group0 = {count/flags, lds_addr, global_addr, type=2},
// group1 = {mask/data_size, tensor dims, tile dims, strides}. n = #f32 elements
// (must be < 65536 to fit tile_dim0; SPAN*FF = 4000 here).
__device__ __forceinline__ void tdm_load_1d(uint32_t lds_off, const void* gptr,
                                            uint32_t n) {
#if defined(USE_TDM)
  uint64_t ga = (uint64_t)(uintptr_t)gptr;
  v4u g0;
  g0.x = 1u;                                             // count=1 (valid), user desc
  g0.y = lds_off;                                        // lds_addr [63:32]
  g0.z = (uint32_t)ga;                                   // global_addr [95:64]
  g0.w = (uint32_t)((ga >> 32) & 0x01FFFFFFu)            // global_addr [120:96]
         | 0x80000000u;                                  // type=2 ("image") [127:126]
  uint32_t td0 = n;                                      // tensor_dim0 (elements)
  uint32_t tile0 = n;                                    // tile_dim0  (elements)
  v8i g1;
  g1[0] = (int)(2u << 16);                               // mask=0, data_size=2 (4B)
  g1[1] = (int)((td0 & 0xFFFFu) << 16);                  // tensor_dim0[15:0]
  g1[2] = (int)((td0 >> 16) | (1u << 16));               // tensor_dim0[31:16], tensor_dim1=1
  g1[3] = (int)((tile0 & 0xFFFFu) << 16);                // tile_dim0
  g1[4] = 1;                                             // tile_dim1=1, tile_dim2=0
  g1[5] = (int)td0;                                      // tensor_dim0_stride lo32
  g1[6] = 0;                                             // stride hi bits = 0
  g1[7] = 0;
  v4i z4 = {0, 0, 0, 0};
#if __clang_major__ >= 23
  v8i z8 = {0, 0, 0, 0, 0, 0, 0, 0};
  __builtin_amdgcn_tensor_load_to_lds(g0, g1, z4, z4, z8, 0);
#else
  __builtin_amdgcn_tensor_load_to_lds(g0, g1, z4, z4, 0);
#endif
#endif
}

// Stage this block's (cf, bw, excitation) tile into LDS.
// Global side: 3 bulk contiguous DMAs; thread side will read via ds_load.
__device__ __forceinline__ void stage_tile(const float* __restrict__ cf,
                                           const float* __restrict__ bw,
                                           const float* __restrict__ ex, int b,
                                           int blockT0, float* s_cf, float* s_bw,
                                           float* s_ex) {
  const size_t cfbase = ((size_t)b * TT + (size_t)blockT0) * FF;
  const size_t exbase = (size_t)b * TT + (size_t)blockT0;
#if defined(USE_TDM)
  // Flat LDS addresses: low 32 bits == LDS byte offset (aperture truncation).
  tdm_load_1d((uint32_t)(uintptr_t)s_cf, cf + cfbase, SPAN * FF);
  tdm_load_1d((uint32_t)(uintptr_t)s_bw, bw + cfbase, SPAN * FF);
  tdm_load_1d((uint32_t)(uintptr_t)s_ex, ex + exbase, SPAN);
  __builtin_amdgcn_s_wait_tensorcnt(0);
  __syncthreads();
#else
  for (int i = threadIdx.x; i < SPAN * FF; i += blockDim.x) {
    s_cf[i] = cf[cfbase + i];
    s_bw[i] = bw[cfbase + i];
  }
  for (int i = threadIdx.x; i < SPAN; i += blockDim.x) s_ex[i] = ex[exbase + i];
  __syncthreads();
#endif
}

// Coefficients for one (t,f) sample.
__device__ __forceinline__ void coeffs(float cfv, float bwv, float& a1, float& a2,
                                       float& g) {
  float r = __expf(-PI_F * bwv * INV_SR);
  float th = TWO_PI_F * cfv * INV_SR;
  float sn, cs;
  __sincosf(th, &sn, &cs);
  a1 = -2.0f * r * cs;
  a2 = r * r;
  g = (1.0f - a2) * sn;
}

// ---------------- Pass 1: per-chunk particular + homogeneous end-states ------
__global__ void formant_pass1(const float* __restrict__ exc,
                              const float* __restrict__ cf,
                              const float* __restrict__ bw,
                              float* __restrict__ sum6) {
  extern __shared__ float smem[];
  float* s_cf = smem;                 // SPAN*FF
  float* s_bw = smem + SPAN * FF;     // SPAN*FF
  float* s_ex = smem + 2 * SPAN * FF; // SPAN
  const int b = blockIdx.y;
  const int blockT0 = blockIdx.x * SPAN;
  stage_tile(cf, bw, exc, b, blockT0, s_cf, s_bw, s_ex);

  const int c = blockIdx.x * WCH + threadIdx.x; // chunk id in [0, CC)
  const int sbase = threadIdx.x * LL;           // local tile offset

  float p1[FF], p2[FF], h11[FF], h12[FF], h21[FF], h22[FF];
#pragma unroll
  for (int f = 0; f < FF; ++f) {
    p1[f] = 0.f; p2[f] = 0.f;       // particular: zero init, driven by x
    h11[f] = 1.f; h12[f] = 0.f;     // homogeneous resp. to (y1,y2)=(1,0)
    h21[f] = 0.f; h22[f] = 1.f;     // homogeneous resp. to (y1,y2)=(0,1)
  }
  for (int i = 0; i < LL; ++i) {
    const int s = sbase + i;
    const float x = s_ex[s];
#pragma unroll
    for (int f = 0; f < FF; ++f) {
      float a1, a2, g;
      coeffs(s_cf[s * FF + f], s_bw[s * FF + f], a1, a2, g);
      float pt = fmaf(-a2, p2[f], fmaf(-a1, p1[f], x * g));
      float h1t = fmaf(-a2, h12[f], -a1 * h11[f]);
      float h2t = fmaf(-a2, h22[f], -a1 * h21[f]);
      p2[f] = p1[f];  p1[f] = pt;
      h12[f] = h11[f]; h11[f] = h1t;
      h22[f] = h21[f]; h21[f] = h2t;
    }
  }
#pragma unroll
  for (int f = 0; f < FF; ++f) {
    float* o = sum6 + (size_t)((b * CC + c) * FF + f) * 6;
    o[0] = p1[f];  o[1] = p2[f];
    o[2] = h11[f]; o[3] = h12[f];
    o[4] = h21[f]; o[5] = h22[f];
  }
}

// ---------------- Pass 2: serial combine across chunks (160 lanes) -----------
__global__ void formant_pass2(const float* __restrict__ states,
                              const float* __restrict__ sum6,
                              float* __restrict__ inits,
                              float* __restrict__ out_states) {
  const int id = blockIdx.x * blockDim.x + threadIdx.x; // id = b*FF + f
  if (id >= BB * FF) return;
  const int b = id / FF, f = id % FF;
  float y1 = states[id * 2 + 0];
  float y2 = states[id * 2 + 1];
  for (int c = 0; c < CC; ++c) {
    const size_t base = (size_t)(b * CC + c) * FF + f;
    inits[base * 2 + 0] = y1;
    inits[base * 2 + 1] = y2;
    const float* s = sum6 + base * 6;
    float n1 = s[0] + y1 * s[2] + y2 * s[4]; // y at chunk's last step
    float n2 = s[1] + y1 * s[3] + y2 * s[5]; // y at second-to-last step
    y1 = n1; y2 = n2;
  }
  out_states[id * 2 + 0] = y1;
  out_states[id * 2 + 1] = y2;
}

// ---------------- Pass 3: replay with true initial states, sum over F --------
__global__ void formant_pass3(const float* __restrict__ exc,
                              const float* __restrict__ cf,
                              const float* __restrict__ bw,
                              const float* __restrict__ inits,
                              float* __restrict__ out) {
  extern __shared__ float smem[];
  float* s_cf = smem;
  float* s_bw = smem + SPAN * FF;
  float* s_ex = smem + 2 * SPAN * FF;
  const int b = blockIdx.y;
  const int blockT0 = blockIdx.x * SPAN;
  stage_tile(cf, bw, exc, b, blockT0, s_cf, s_bw, s_ex);

  const int c = blockIdx.x * WCH + threadIdx.x;
  const int sbase = threadIdx.x * LL;

  float y1[FF], y2[FF];
#pragma unroll
  for (int f = 0; f < FF; ++f) {
    const size_t base = (size_t)(b * CC + c) * FF + f;
    y1[f] = inits[base * 2 + 0];
    y2[f] = inits[base * 2 + 1];
  }
  float* op = out + (size_t)b * TT + (size_t)c * LL;
  for (int i = 0; i < LL; ++i) {
    const int s = sbase + i;
    const float x = s_ex[s];
    float acc = 0.f;
#pragma unroll
    for (int f = 0; f < FF; ++f) {
      float a1, a2, g;
      coeffs(s_cf[s * FF + f], s_bw[s * FF + f], a1, a2, g);
      float yt = fmaf(-a2, y2[f], fmaf(-a1, y1[f], x * g));
      y2[f] = y1[f];
      y1[f] = yt;
      acc += yt;
    }
    op[i] = acc;
  }
}

// ---------------- Fallback: fully serial (if workspace too small) ------------
__global__ void formant_serial(const float* __restrict__ exc,
                               const float* __restrict__ cf,
                               const float* __restrict__ bw,
                               const float* __restrict__ states,
                               float* __restrict__ out,
                               float* __restrict__ out_states) {
  const int b = blockIdx.x * blockDim.x + threadIdx.x;
  if (b >= BB) return;
  float y1[FF], y2[FF];
#pragma unroll
  for (int f = 0; f < FF; ++f) {
    y1[f] = states[(b * FF + f) * 2 + 0];
    y2[f] = states[(b * FF + f) * 2 + 1];
  }
  for (int t = 0; t < TT; ++t) {
    const size_t e = (size_t)b * TT + t;
    const float x = exc[e];
    float acc = 0.f;
#pragma unroll
    for (int f = 0; f < FF; ++f) {
      float a1, a2, g;
      coeffs(cf[e * FF + f], bw[e * FF + f], a1, a2, g);
      float yt = fmaf(-a2, y2[f], fmaf(-a1, y1[f], x * g));
      y2[f] = y1[f];
      y1[f] = yt;
      acc += yt;
    }
    out[e] = acc;
  }
#pragma unroll
  for (int f = 0; f < FF; ++f) {
    out_states[(b * FF + f) * 2 + 0] = y1[f];
    out_states[(b * FF + f) * 2 + 1] = y2[f];
  }
}

// ---------------- host launch ----------------
extern "C" void kernel_launch(void* const* d_in, const int* in_sizes, int n_in,
                              void* d_out, int out_size, void* d_ws,
                              size_t ws_size, hipStream_t stream) {
  const float* exc = (const float*)d_in[0];    // (B,T,1)
  const float* cf = (const float*)d_in[1];     // (B,T,F)
  const float* bw = (const float*)d_in[2];     // (B,T,F)
  const float* states = (const float*)d_in[3]; // (B,F,2)
  float* out = (float*)d_out;                  // (B,T,1) then (B,F,2)
  float* out_states = out + (size_t)BB * TT;

  const size_t need_sum6 = (size_t)BB * CC * FF * 6 * sizeof(float);
  const size_t need_init = (size_t)BB * CC * FF * 2 * sizeof(float);
  if (ws_size < need_sum6 + need_init) {
    formant_serial<<<1, BB, 0, stream>>>(exc, cf, bw, states, out, out_states);
    return;
  }
  float* sum6 = (float*)d_ws;
  float* inits = sum6 + (size_t)BB * CC * FF * 6;

  const size_t shmem = (size_t)(2 * SPAN * FF + SPAN) * sizeof(float); // 35,200 B
  formant_pass1<<<dim3(NBLK, BB), WCH, shmem, stream>>>(exc, cf, bw, sum6);
  formant_pass2<<<1, 192, 0, stream>>>(states, sum6, inits, out_states);
  formant_pass3<<<dim3(NBLK, BB), WCH, shmem, stream>>>(exc, cf, bw, inits, out);
}